// OSNAP_85710367359546
// MI455X (gfx1250) — compile-verified
//
#include <hip/hip_runtime.h>

// OSNAP sketch: out[n,f] = sum_d x[n,d] * P[f,d], P has exactly 4 nonzeros
// (+-0.5) per column d. Bandwidth-bound => sparse two-phase algorithm.

#define D_IN   4096
#define D_F    8192
#define CAP    32      // padded ELL capacity per output feature (avg nnz = 2)
#define ROWS   4       // x rows staged in LDS per block (4*4096*4B = 64 KB)
#define TPB    256

#define AS_GLOBAL __attribute__((address_space(1)))
#define AS_LDS    __attribute__((address_space(3)))

typedef int v4i __attribute__((vector_size(16)));

static __device__ __forceinline__ unsigned ballot32(bool pred) {
#if __has_builtin(__builtin_amdgcn_ballot_w32)
  return __builtin_amdgcn_ballot_w32(pred);
#else
  return (unsigned)__ballot(pred);
#endif
}

// CDNA5 async data-mover: global -> LDS, 16B per lane, tracked by ASYNCcnt.
static __device__ __forceinline__ void async_copy_b128(const float* gsrc, float* ldst) {
#if __has_builtin(__builtin_amdgcn_global_load_async_to_lds_b128)
  __builtin_amdgcn_global_load_async_to_lds_b128(
      (AS_GLOBAL v4i*)gsrc, (AS_LDS v4i*)ldst, 0, 0);
#else
  unsigned loff = (unsigned)(unsigned long long)ldst;  // low 32 bits = LDS offset
  asm volatile("global_load_async_to_lds_b128 %0, %1, off"
               :: "v"(loff), "v"(gsrc) : "memory");
#endif
}

static __device__ __forceinline__ void async_wait0() {
#if __has_builtin(__builtin_amdgcn_s_wait_asynccnt)
  __builtin_amdgcn_s_wait_asynccnt(0);
#else
  asm volatile("s_wait_asynccnt 0" ::: "memory");
#endif
}

// ---------------------------------------------------------------------------
// Phase 2 (defined first so the disasm snippet shows it): gather. Each block
// owns ROWS consecutive x rows: async-stage them into LDS (64 KB) with a
// straight-line burst of 16 async b128 transfers per lane, then each thread
// produces output features f = tid + k*TPB for all ROWS rows (ELL metadata
// read once per block from L2, reused ROWS x). Output stores are fully
// coalesced and non-temporal (write-once stream, keep L2 for metadata).
// ---------------------------------------------------------------------------
__global__ __launch_bounds__(TPB) void osnap_gather(
    const float* __restrict__ x,
    const int* __restrict__ cntF, const int2* __restrict__ pairF,
    float* __restrict__ out, int N) {
  __shared__ float xs[ROWS * D_IN];

  const int tid = threadIdx.x;
  // Clamp base row so the block's ROWS rows are always fully in range
  // (exact when N % ROWS == 0, which holds for N = 16384).
  int n0 = blockIdx.x * ROWS;
  if (n0 + ROWS > N) n0 = (N >= ROWS) ? (N - ROWS) : 0;
  const float* gx = x + (size_t)n0 * D_IN;

  // Stage ROWS*D_IN floats: 16 unguarded async ops per lane, one wait.
#pragma unroll
  for (int k = 0; k < (ROWS * D_IN) / (TPB * 4); ++k) {
    const int i = tid * 4 + k * (TPB * 4);
    async_copy_b128(gx + i, &xs[i]);
  }
  async_wait0();
  __syncthreads();

  for (int f = tid; f < D_F; f += TPB) {
    int cnt = cntF[f];
    if (cnt > CAP) cnt = CAP;
    const int2* pp = pairF + (size_t)f * CAP;

    float a0 = 0.f, a1 = 0.f, a2 = 0.f, a3 = 0.f;
    for (int j = 0; j < cnt; ++j) {
      int2 pr = pp[j];                   // one b64 load per nnz
      int d   = pr.x;
      float v = __int_as_float(pr.y);
      a0 = fmaf(xs[d],            v, a0);
      a1 = fmaf(xs[D_IN + d],     v, a1);
      a2 = fmaf(xs[2 * D_IN + d], v, a2);
      a3 = fmaf(xs[3 * D_IN + d], v, a3);
    }
    size_t o = (size_t)n0 * D_F + f;
    __builtin_nontemporal_store(a0, &out[o]);
    __builtin_nontemporal_store(a1, &out[o + (size_t)D_F]);
    __builtin_nontemporal_store(a2, &out[o + 2 * (size_t)D_F]);
    __builtin_nontemporal_store(a3, &out[o + 3 * (size_t)D_F]);
  }
}

// ---------------------------------------------------------------------------
// Phase 1: extract per-feature ELL from dense P. One wave per feature row f
// scans its 4096 contiguous floats (coalesced across 32 lanes, non-temporal:
// P is read exactly once), compacts nonzeros with ballot + prefix popcount.
// Deterministic (ordered by d), no atomics.
// ---------------------------------------------------------------------------
__global__ __launch_bounds__(TPB) void osnap_build(
    const float* __restrict__ P,
    int* __restrict__ cntF, int2* __restrict__ pairF) {
  const int wavesPerBlock = TPB / 32;
  const int wave = threadIdx.x >> 5;
  const int lane = threadIdx.x & 31;
  const int f = blockIdx.x * wavesPerBlock + wave;
  if (f >= D_F) return;

  const float* row = P + (size_t)f * D_IN;
  int cnt = 0;
  for (int base = 0; base < D_IN; base += 32) {
    float p = __builtin_nontemporal_load(&row[base + lane]);
    unsigned mask = ballot32(p != 0.0f);
    if (p != 0.0f) {
      int pos = cnt + __popc(mask & ((1u << lane) - 1u));
      if (pos < CAP) {
        int2 pr;
        pr.x = base + lane;
        pr.y = __float_as_int(p);
        pairF[(size_t)f * CAP + pos] = pr;
      }
    }
    cnt += __popc(mask);
  }
  if (lane == 0) cntF[f] = (cnt < CAP) ? cnt : CAP;
}

extern "C" void kernel_launch(void* const* d_in, const int* in_sizes, int n_in,
                              void* d_out, int out_size, void* d_ws, size_t ws_size,
                              hipStream_t stream) {
  const float* x = (const float*)d_in[0];   // [N, 4096] f32
  const float* P = (const float*)d_in[1];   // [8192, 4096] f32
  float* out = (float*)d_out;               // [N, 8192] f32
  const int N = in_sizes[0] / D_IN;

  // Workspace layout: cntF[8192] | pairF[8192*CAP] int2   (~2.03 MB)
  char* ws = (char*)d_ws;
  int*  cntF  = (int*)ws;
  int2* pairF = (int2*)(ws + (size_t)D_F * sizeof(int));

  const int buildBlocks = D_F / (TPB / 32);           // 1024 (one wave per f)
  osnap_build<<<buildBlocks, TPB, 0, stream>>>(P, cntF, pairF);

  const int gatherBlocks = (N + ROWS - 1) / ROWS;     // 4096
  osnap_gather<<<gatherBlocks, TPB, 0, stream>>>(x, cntF, pairF, out, N);
}